// APPNPbase_9938554323113
// MI455X (gfx1250) — compile-verified
//
#include <hip/hip_runtime.h>
#include <hip/hip_bf16.h>
#include <math.h>

typedef __attribute__((ext_vector_type(2))) float v2f;
typedef __attribute__((ext_vector_type(8))) float v8f;

#define NNODES   100000
#define IN_DIM   512
#define H_DIM    64
#define C_OUT    40
#define KSTEPS   10
#define ALPHA    0.1f
#define BETA     0.9f        // 1 - alpha

#define XS_STRIDE 516        // 512 + 4 pad: stride%64 == 4 -> conflict-free A reads
#define HS_STRIDE 68         // 64 + 4 pad

// ---------------------------------------------------------------------------
// Encoder: h = relu(x @ W1 + b1) @ W2 + b2   via V_WMMA_F32_16X16X4_F32
// One wave per 16-row tile. N = 100000 is divisible by 16 (6250 tiles).
// ---------------------------------------------------------------------------
__global__ __launch_bounds__(32)
void encoder_kernel(const float* __restrict__ x,
                    const float* __restrict__ W1, const float* __restrict__ b1,
                    const float* __restrict__ W2, const float* __restrict__ b2,
                    float* __restrict__ h_out)
{
    __shared__ float smem[16 * XS_STRIDE];

    const int lane = threadIdx.x;          // 0..31 (wave32)
    const int row0 = blockIdx.x * 16;

    // ---- stage 16x512 x-tile into LDS (coalesced float4, padded rows) ----
    {
        const float4* __restrict__ xsrc = (const float4*)(x + (size_t)row0 * IN_DIM);
        float4* sdst = (float4*)smem;      // XS_STRIDE/4 = 129 float4 per row
        for (int i = lane; i < 16 * (IN_DIM / 4); i += 32) {
            int r = i >> 7;                // / 128
            int j = i & 127;
            sdst[r * (XS_STRIDE / 4) + j] = xsrc[r * (IN_DIM / 4) + j];
        }
    }
    __syncthreads();

    const int arow = lane & 15;            // A-matrix row (M)
    const int kg   = (lane >> 4) * 2;      // K sub-offset: lanes 0-15 -> K0/K1, 16-31 -> K2/K3
    const int coll = lane & 15;            // B/C column within 16-wide tile

    // ---- GEMM1: 16x512 @ 512x64 -> 16x64 (4 column tiles) ----
    v8f acc[4];
    #pragma unroll
    for (int n = 0; n < 4; ++n) acc[n] = (v8f)0.0f;

    for (int k = 0; k < IN_DIM; k += 4) {
        v2f av = *(const v2f*)&smem[arow * XS_STRIDE + k + kg];
        #pragma unroll
        for (int n = 0; n < 4; ++n) {
            const float* wp = W1 + (size_t)(k + kg) * H_DIM + n * 16 + coll;
            v2f bv;
            bv.x = wp[0];
            bv.y = wp[H_DIM];
            acc[n] = __builtin_amdgcn_wmma_f32_16x16x4_f32(
                false, av, false, bv, (short)0, acc[n], false, false);
        }
    }

    // ---- bias + ReLU (C layout: VGPR i -> row i + 8*(lane>=16), col = coll) ----
    #pragma unroll
    for (int n = 0; n < 4; ++n) {
        float bias = b1[n * 16 + coll];
        #pragma unroll
        for (int i = 0; i < 8; ++i) {
            float v = acc[n][i] + bias;
            acc[n][i] = v > 0.0f ? v : 0.0f;
        }
    }

    // ---- write h1 (16x64) back to LDS in row-major A-usable form ----
    __syncthreads();
    #pragma unroll
    for (int n = 0; n < 4; ++n) {
        #pragma unroll
        for (int i = 0; i < 8; ++i) {
            int r = i + (lane >> 4) * 8;
            smem[r * HS_STRIDE + n * 16 + coll] = acc[n][i];
        }
    }
    __syncthreads();

    // ---- GEMM2: 16x64 @ 64x40 -> 16x40 (3 column tiles, last partially valid) ----
    v8f acc2[3];
    #pragma unroll
    for (int n = 0; n < 3; ++n) acc2[n] = (v8f)0.0f;

    for (int k = 0; k < H_DIM; k += 4) {
        v2f av = *(const v2f*)&smem[arow * HS_STRIDE + k + kg];
        #pragma unroll
        for (int n = 0; n < 3; ++n) {
            int col = n * 16 + coll;
            v2f bv;
            bv.x = (col < C_OUT) ? W2[(size_t)(k + kg) * C_OUT + col] : 0.0f;
            bv.y = (col < C_OUT) ? W2[(size_t)(k + kg + 1) * C_OUT + col] : 0.0f;
            acc2[n] = __builtin_amdgcn_wmma_f32_16x16x4_f32(
                false, av, false, bv, (short)0, acc2[n], false, false);
        }
    }

    // ---- bias b2 + store h[16 x 40] ----
    #pragma unroll
    for (int n = 0; n < 3; ++n) {
        int col = n * 16 + coll;
        if (col < C_OUT) {
            float bias = b2[col];
            #pragma unroll
            for (int i = 0; i < 8; ++i) {
                int r = i + (lane >> 4) * 8;
                h_out[(size_t)(row0 + r) * C_OUT + col] = acc2[n][i] + bias;
            }
        }
    }
}

// ---------------------------------------------------------------------------
// Degree / normalization kernels
// ---------------------------------------------------------------------------
__global__ void deg_init_kernel(float* __restrict__ deg, int n)
{
    int i = blockIdx.x * blockDim.x + threadIdx.x;
    if (i < n) deg[i] = 1.0f;   // self-loop contribution
}

__global__ void deg_accum_kernel(const int* __restrict__ cols,
                                 float* __restrict__ deg, int E)
{
    int e = blockIdx.x * blockDim.x + threadIdx.x;
    if (e < E) unsafeAtomicAdd(&deg[cols[e]], 1.0f);
}

__global__ void dinv_kernel(float* __restrict__ deg_dinv, int n)
{
    int i = blockIdx.x * blockDim.x + threadIdx.x;
    if (i < n) {
        float d = deg_dinv[i];
        deg_dinv[i] = d > 0.0f ? rsqrtf(d) : 0.0f;
    }
}

__global__ void edge_norm_kernel(const int* __restrict__ rows,
                                 const int* __restrict__ cols,
                                 const float* __restrict__ dinv,
                                 float* __restrict__ norm, int E)
{
    int e = blockIdx.x * blockDim.x + threadIdx.x;
    if (e < E) norm[e] = dinv[rows[e]] * dinv[cols[e]];
}

// ---------------------------------------------------------------------------
// APPNP propagation step: z_out = beta*(A_hat z_in) + alpha*h
//   init: self-loop term + alpha*h ;  scatter: edge contributions (atomic)
// ---------------------------------------------------------------------------
__global__ void prop_init_kernel(const float* __restrict__ z_in,
                                 const float* __restrict__ h,
                                 const float* __restrict__ dinv,
                                 float* __restrict__ z_out, int n)
{
    int gid = blockIdx.x * blockDim.x + threadIdx.x;
    if (gid < n * C_OUT) {
        int i = gid / C_OUT;
        float di = dinv[i];
        z_out[gid] = BETA * (di * di) * z_in[gid] + ALPHA * h[gid];
    }
}

__global__ void prop_scatter_kernel(const float* __restrict__ z_in,
                                    const int* __restrict__ rows,
                                    const int* __restrict__ cols,
                                    const float* __restrict__ norm,
                                    float* __restrict__ z_out, int E)
{
    int gid = blockIdx.x * blockDim.x + threadIdx.x;   // E*8 = 25.6M fits in int
    if (gid >= E * 8) return;
    int e  = gid >> 3;
    int c0 = (gid & 7) * 5;                            // 8 lanes x 5 channels = 40
    int r = rows[e];
    int c = cols[e];
    float w = BETA * norm[e];
    const float* zp = z_in + (size_t)r * C_OUT + c0;
    float*       op = z_out + (size_t)c * C_OUT + c0;
    #pragma unroll
    for (int j = 0; j < 5; ++j)
        unsafeAtomicAdd(op + j, w * zp[j]);
}

// ---------------------------------------------------------------------------
// log_softmax over C_OUT=40 classes, one thread per row
// ---------------------------------------------------------------------------
__global__ void log_softmax_kernel(const float* __restrict__ z,
                                   float* __restrict__ out, int n)
{
    int i = blockIdx.x * blockDim.x + threadIdx.x;
    if (i >= n) return;
    const float* zp = z + (size_t)i * C_OUT;
    float v[C_OUT];
    float m = -3.4e38f;
    #pragma unroll
    for (int c = 0; c < C_OUT; ++c) { v[c] = zp[c]; m = fmaxf(m, v[c]); }
    float s = 0.0f;
    #pragma unroll
    for (int c = 0; c < C_OUT; ++c) s += expf(v[c] - m);
    float ls = m + logf(s);
    float* op = out + (size_t)i * C_OUT;
    #pragma unroll
    for (int c = 0; c < C_OUT; ++c) op[c] = v[c] - ls;
}

// ---------------------------------------------------------------------------
extern "C" void kernel_launch(void* const* d_in, const int* in_sizes, int n_in,
                              void* d_out, int out_size, void* d_ws, size_t ws_size,
                              hipStream_t stream)
{
    const float* x  = (const float*)d_in[0];
    const int*   ei = (const int*)d_in[1];
    const float* W1 = (const float*)d_in[2];
    const float* b1 = (const float*)d_in[3];
    const float* W2 = (const float*)d_in[4];
    const float* b2 = (const float*)d_in[5];

    const int E = in_sizes[1] / 2;
    const int* rows = ei;        // edge_index[0] = source
    const int* cols = ei + E;    // edge_index[1] = destination

    float* out = (float*)d_out;

    // workspace carve-up (floats): h | z0 | z1 | dinv | norm  (~61 MB)
    float* ws   = (float*)d_ws;
    float* h    = ws;
    float* z0   = h    + (size_t)NNODES * C_OUT;
    float* z1   = z0   + (size_t)NNODES * C_OUT;
    float* dinv = z1   + (size_t)NNODES * C_OUT;
    float* norm = dinv + (size_t)NNODES;

    const int T = 256;

    // encoder (WMMA)
    encoder_kernel<<<NNODES / 16, 32, 0, stream>>>(x, W1, b1, W2, b2, h);

    // GCN normalization
    deg_init_kernel<<<(NNODES + T - 1) / T, T, 0, stream>>>(dinv, NNODES);
    deg_accum_kernel<<<(E + T - 1) / T, T, 0, stream>>>(cols, dinv, E);
    dinv_kernel<<<(NNODES + T - 1) / T, T, 0, stream>>>(dinv, NNODES);
    edge_norm_kernel<<<(E + T - 1) / T, T, 0, stream>>>(rows, cols, dinv, norm, E);

    // K propagation steps, ping-pong between z0/z1 (z starts as h)
    const float* zin = h;
    for (int s = 0; s < KSTEPS; ++s) {
        float* zout = (s & 1) ? z1 : z0;
        prop_init_kernel<<<(NNODES * C_OUT + T - 1) / T, T, 0, stream>>>(zin, h, dinv, zout, NNODES);
        prop_scatter_kernel<<<((E * 8) + T - 1) / T, T, 0, stream>>>(zin, rows, cols, norm, zout, E);
        zin = zout;
    }

    log_softmax_kernel<<<(NNODES + T - 1) / T, T, 0, stream>>>(zin, out, NNODES);
}